// GCNDiscriminator_9191230014152
// MI455X (gfx1250) — compile-verified
//
#include <hip/hip_runtime.h>
#include <math.h>

typedef __attribute__((ext_vector_type(2))) float v2f;
typedef __attribute__((ext_vector_type(8))) float v8f;

#define N_NODES 50000
#define N_EDGES 800000
#define DIM     256
#define MTILES  (N_NODES / 16)   // 3125, exact

// Workspace layout in floats:
//   [0, 50000)                    deg -> dinv (rsqrt in place)
//   [HS_OFF, HS_OFF+12.8M)        hs  = (x@W) * dinv[row]
//   [ACC_OFF, ACC_OFF+12.8M)      acc = hs (self loop) + scatter of hs[row]
//   [CS_OFF, CS_OFF+256)          column sums of sigmoid(agg)
#define HS_OFF  50176
#define ACC_OFF (HS_OFF + N_NODES * DIM)
#define CS_OFF  (ACC_OFF + N_NODES * DIM)

// ---------------- degree / normalization ----------------

__global__ void init_deg_kernel(float* __restrict__ deg, float* __restrict__ colsum) {
    int i = blockIdx.x * blockDim.x + threadIdx.x;
    if (i < N_NODES) deg[i] = 1.0f;          // self loop
    if (i < DIM)     colsum[i] = 0.0f;       // zero reduction buffer each call
}

__global__ void count_deg_kernel(const long long* __restrict__ ei, float* __restrict__ deg) {
    int e = blockIdx.x * blockDim.x + threadIdx.x;
    if (e >= N_EDGES) return;
    int c = (int)ei[N_EDGES + e];            // col = target
    atomicAdd(&deg[c], 1.0f);
}

__global__ void rsqrt_deg_kernel(float* __restrict__ deg) {
    int i = blockIdx.x * blockDim.x + threadIdx.x;
    if (i < N_NODES) deg[i] = rsqrtf(deg[i]);
}

// ---------------- GEMM: hs = (x @ W) * dinv[row], acc = hs ----------------
// One wave computes a 16-row x 64-col strip using V_WMMA_F32_16X16X4_F32.

__global__ void __launch_bounds__(256)
gemm_wmma_kernel(const float* __restrict__ x,
                 const float* __restrict__ W,
                 const float* __restrict__ dinv,
                 float* __restrict__ hs,
                 float* __restrict__ acc) {
    const int lane = threadIdx.x & 31;
    const int wave = threadIdx.x >> 5;
    const int w    = blockIdx.x * 8 + wave;       // global wave id
    const int tileM = w >> 2;                     // 16-row strip
    if (tileM >= MTILES) return;                  // wave-uniform: EXEC stays all-1s below
    const int m0 = tileM * 16;
    const int n0 = (w & 3) * 64;                  // 64-column quadrant
    const int hi = lane >> 4;                     // 0 or 1
    const int lo = lane & 15;

    v8f accv0 = {}, accv1 = {}, accv2 = {}, accv3 = {};

    const float* arow = x + (size_t)(m0 + lo) * DIM + 2 * hi;

    for (int k = 0; k < DIM; k += 4) {
        v2f a;
        a.x = arow[k];
        a.y = arow[k + 1];
        const float* bcol = W + (size_t)(k + 2 * hi) * DIM + n0 + lo;

        v2f b0; b0.x = bcol[0];      b0.y = bcol[0 + DIM];
        v2f b1; b1.x = bcol[16];     b1.y = bcol[16 + DIM];
        v2f b2; b2.x = bcol[32];     b2.y = bcol[32 + DIM];
        v2f b3; b3.x = bcol[48];     b3.y = bcol[48 + DIM];

        accv0 = __builtin_amdgcn_wmma_f32_16x16x4_f32(false, a, false, b0, (short)0, accv0, false, false);
        accv1 = __builtin_amdgcn_wmma_f32_16x16x4_f32(false, a, false, b1, (short)0, accv1, false, false);
        accv2 = __builtin_amdgcn_wmma_f32_16x16x4_f32(false, a, false, b2, (short)0, accv2, false, false);
        accv3 = __builtin_amdgcn_wmma_f32_16x16x4_f32(false, a, false, b3, (short)0, accv3, false, false);
    }

    // Epilogue: scale by dinv[m], write hs and acc (= self-loop init).
#pragma unroll
    for (int j = 0; j < 8; ++j) {
        const int m = m0 + j + 8 * hi;
        const float s = dinv[m];
        const size_t base = (size_t)m * DIM + n0 + lo;
        float v0 = accv0[j] * s;
        float v1 = accv1[j] * s;
        float v2 = accv2[j] * s;
        float v3 = accv3[j] * s;
        hs[base + 0]  = v0;  acc[base + 0]  = v0;
        hs[base + 16] = v1;  acc[base + 16] = v1;
        hs[base + 32] = v2;  acc[base + 32] = v2;
        hs[base + 48] = v3;  acc[base + 48] = v3;
    }
}

// ---------------- edge scatter: acc[col] += hs[row] ----------------
// 4 edges per 256-thread block; 64 threads per edge, float4 each.

__global__ void __launch_bounds__(256)
scatter_kernel(const long long* __restrict__ ei,
               const float* __restrict__ hs,
               float* __restrict__ acc) {
    const int e = blockIdx.x * 4 + (threadIdx.x >> 6);
    if (e >= N_EDGES) return;
    const int g = (threadIdx.x & 63) * 4;
    const long long r = ei[e];
    const long long c = ei[N_EDGES + e];
    const float4 v = *(const float4*)(hs + (size_t)r * DIM + g);
    float* dst = acc + (size_t)c * DIM + g;
    atomicAdd(dst + 0, v.x);
    atomicAdd(dst + 1, v.y);
    atomicAdd(dst + 2, v.z);
    atomicAdd(dst + 3, v.w);
}

// ---------------- sigmoid + column-sum ----------------

__global__ void __launch_bounds__(256)
sigmoid_colsum_kernel(const float* __restrict__ acc,
                      const float* __restrict__ dinv,
                      const float* __restrict__ b_conv,
                      float* __restrict__ colsum) {
    const int t = threadIdx.x;                 // column, 0..255
    const int r0 = blockIdx.x * 64;
    const float b = b_conv[t];
    int rend = r0 + 64; if (rend > N_NODES) rend = N_NODES;
    float sum = 0.0f;
    for (int r = r0; r < rend; ++r) {
        float a = acc[(size_t)r * DIM + t] * dinv[r] + b;
        sum += 1.0f / (1.0f + __expf(-a));
    }
    atomicAdd(&colsum[t], sum);
}

// ---------------- final: mean -> dot w_lin -> sigmoid ----------------

__global__ void final_kernel(const float* __restrict__ colsum,
                             const float* __restrict__ w_lin,
                             const float* __restrict__ b_lin,
                             float* __restrict__ out) {
    __shared__ float red[256];
    const int t = threadIdx.x;
    red[t] = colsum[t] * (1.0f / (float)N_NODES) * w_lin[t];
    __syncthreads();
    for (int s = 128; s > 0; s >>= 1) {
        if (t < s) red[t] += red[t + s];
        __syncthreads();
    }
    if (t == 0) out[0] = 1.0f / (1.0f + __expf(-(red[0] + b_lin[0])));
}

// ---------------- launch ----------------

extern "C" void kernel_launch(void* const* d_in, const int* in_sizes, int n_in,
                              void* d_out, int out_size, void* d_ws, size_t ws_size,
                              hipStream_t stream) {
    const float*     x      = (const float*)d_in[0];
    const long long* ei     = (const long long*)d_in[1];
    const float*     W      = (const float*)d_in[2];
    const float*     b_conv = (const float*)d_in[3];
    const float*     w_lin  = (const float*)d_in[4];
    const float*     b_lin  = (const float*)d_in[5];
    float* out = (float*)d_out;

    float* ws     = (float*)d_ws;
    float* deg    = ws;             // becomes dinv after rsqrt
    float* hs     = ws + HS_OFF;
    float* acc    = ws + ACC_OFF;
    float* colsum = ws + CS_OFF;

    init_deg_kernel<<<(N_NODES + 255) / 256, 256, 0, stream>>>(deg, colsum);
    count_deg_kernel<<<(N_EDGES + 255) / 256, 256, 0, stream>>>(ei, deg);
    rsqrt_deg_kernel<<<(N_NODES + 255) / 256, 256, 0, stream>>>(deg);

    // 3125 M-tiles * 4 column-quadrant waves = 12500 waves, 8 waves/block
    gemm_wmma_kernel<<<(MTILES * 4 + 7) / 8, 256, 0, stream>>>(x, W, deg, hs, acc);

    scatter_kernel<<<(N_EDGES + 3) / 4, 256, 0, stream>>>(ei, hs, acc);

    sigmoid_colsum_kernel<<<(N_NODES + 63) / 64, 256, 0, stream>>>(acc, deg, b_conv, colsum);

    final_kernel<<<1, 256, 0, stream>>>(colsum, w_lin, b_lin, out);
}